// TensorProductReference_54254026883874
// MI455X (gfx1250) — compile-verified
//
#include <hip/hip_runtime.h>

#if defined(__HIP_DEVICE_COMPILE__) && !__has_builtin(__builtin_amdgcn_wmma_f32_16x16x4_f32)
#error "builtin __builtin_amdgcn_wmma_f32_16x16x4_f32 not available for this device target"
#endif

typedef float v2f __attribute__((ext_vector_type(2)));
typedef float v4f __attribute__((ext_vector_type(4)));
typedef float v8f __attribute__((ext_vector_type(8)));

namespace tp {

constexpr int NPATH = 23;
constexpr int KOUT  = 99;    // total output columns
constexpr int KPAD  = 112;   // padded to 7 tiles of 16
constexpr int NCH   = 32;
constexpr int XD    = 16;

// _INS enumeration order from the reference (verified by static_assert below)
constexpr int PATH_L[NPATH][3] = {
  {0,0,0},{0,1,1},{0,2,2},{0,3,3},
  {1,0,1},{1,1,0},{1,1,2},{1,2,1},{1,2,3},{1,3,2},
  {2,0,2},{2,1,1},{2,1,3},{2,2,0},{2,2,2},{2,3,1},{2,3,3},
  {3,0,3},{3,1,2},{3,2,1},{3,2,3},{3,3,0},{3,3,2}
};

constexpr bool checkPaths() {
  int c = 0;
  for (int l1 = 0; l1 <= 3; ++l1)
    for (int l2 = 0; l2 <= 3; ++l2) {
      int lo = l1 > l2 ? l1 - l2 : l2 - l1;
      int hi = (l1 + l2) < 3 ? (l1 + l2) : 3;
      for (int l3 = lo; l3 <= hi; ++l3) {
        if ((l1 + l2 + l3) % 2) continue;
        if (c >= NPATH) return false;
        if (PATH_L[c][0] != l1 || PATH_L[c][1] != l2 || PATH_L[c][2] != l3) return false;
        ++c;
      }
    }
  return c == NPATH;
}
static_assert(checkPaths(), "path enumeration mismatch");

constexpr int OFFS[5] = {0, 1, 4, 9, 16};

constexpr double FACT[11] = {1., 1., 2., 6., 24., 120., 720., 5040., 40320., 362880., 3628800.};

constexpr double csqrt(double x) {
  if (x <= 0.0) return 0.0;
  double g = x > 1.0 ? x : 1.0;
  for (int i = 0; i < 48; ++i) g = 0.5 * (g + x / g);
  return g;
}

struct Cx   { double re, im; };
struct Arr3 { double v[7][7][7]; };
struct QMat { Cx q[7][7]; };
struct W3   { double v[7][7][7]; };

constexpr Arr3 su2_cg(int j1, int j2, int j3) {
  Arr3 out{};
  for (int m1 = -j1; m1 <= j1; ++m1)
    for (int m2 = -j2; m2 <= j2; ++m2) {
      int m3 = m1 + m2;
      if (m3 < -j3 || m3 > j3) continue;
      int vmin = 0;
      if (-j1 + j2 + m3 > vmin) vmin = -j1 + j2 + m3;
      if (-j1 + m1 > vmin)      vmin = -j1 + m1;
      int vmax = j2 + j3 + m1;
      if (j3 - j1 + j2 < vmax) vmax = j3 - j1 + j2;
      if (j3 + m3 < vmax)      vmax = j3 + m3;
      double Cc = csqrt((2.0 * j3 + 1.0) * FACT[j3 + j1 - j2] * FACT[j3 - j1 + j2] *
                        FACT[j1 + j2 - j3] / FACT[j1 + j2 + j3 + 1]);
      Cc *= csqrt(FACT[j3 + m3] * FACT[j3 - m3] * FACT[j1 - m1] * FACT[j1 + m1] *
                  FACT[j2 - m2] * FACT[j2 + m2]);
      double S = 0.0;
      for (int v = vmin; v <= vmax; ++v) {
        double sg = ((((v + j2 + m2) % 2) + 2) % 2 == 0) ? 1.0 : -1.0;
        S += sg * FACT[j2 + j3 + m1 - v] * FACT[j1 - m1 + v] /
             (FACT[v] * FACT[j3 - j1 + j2 - v] * FACT[j3 + m3 - v] * FACT[v + j1 - j2 - m3]);
      }
      out.v[m1 + j1][m2 + j2][m3 + j3] = Cc * S;
    }
  return out;
}

constexpr QMat qmat(int l) {
  QMat Q{};
  const double s = 1.0 / csqrt(2.0);
  for (int m = -l; m < 0; ++m) {
    Q.q[l + m][l - m] = Cx{s, 0.0};    // q[l+m, l+|m|]
    Q.q[l + m][l + m] = Cx{0.0, -s};   // q[l+m, l-|m|]
  }
  Q.q[l][l] = Cx{1.0, 0.0};
  for (int m = 1; m <= l; ++m) {
    double sg = (m % 2 == 0) ? s : -s;
    Q.q[l + m][l + m] = Cx{sg, 0.0};
    Q.q[l + m][l - m] = Cx{0.0, sg};
  }
  // multiply by (-i)^l
  Cx ph = (l % 4 == 0) ? Cx{1, 0} : (l % 4 == 1) ? Cx{0, -1} : (l % 4 == 2) ? Cx{-1, 0} : Cx{0, 1};
  for (int a = 0; a < 2 * l + 1; ++a)
    for (int b = 0; b < 2 * l + 1; ++b) {
      Cx v = Q.q[a][b];
      Q.q[a][b] = Cx{v.re * ph.re - v.im * ph.im, v.re * ph.im + v.im * ph.re};
    }
  return Q;
}

constexpr W3 wigner3j(int l1, int l2, int l3) {
  const int n1 = 2 * l1 + 1, n2 = 2 * l2 + 1, n3 = 2 * l3 + 1;
  Arr3 C  = su2_cg(l1, l2, l3);
  QMat Q1 = qmat(l1), Q2 = qmat(l2), Q3 = qmat(l3);
  // staged einsum('ij,kl,mn,ikn->jlm', Q1, Q2, conj(Q3.T), C)
  Cx A1[7][7][7]{};
  for (int j = 0; j < n1; ++j)
    for (int k = 0; k < n2; ++k)
      for (int n = 0; n < n3; ++n) {
        Cx s{0, 0};
        for (int i = 0; i < n1; ++i) {
          s.re += Q1.q[i][j].re * C.v[i][k][n];
          s.im += Q1.q[i][j].im * C.v[i][k][n];
        }
        A1[j][k][n] = s;
      }
  Cx A2[7][7][7]{};
  for (int j = 0; j < n1; ++j)
    for (int l = 0; l < n2; ++l)
      for (int n = 0; n < n3; ++n) {
        Cx s{0, 0};
        for (int k = 0; k < n2; ++k) {
          Cx a = Q2.q[k][l], b = A1[j][k][n];
          s.re += a.re * b.re - a.im * b.im;
          s.im += a.re * b.im + a.im * b.re;
        }
        A2[j][l][n] = s;
      }
  W3 out{};
  double nrm = 0.0;
  for (int j = 0; j < n1; ++j)
    for (int l = 0; l < n2; ++l)
      for (int m = 0; m < n3; ++m) {
        double re = 0.0;
        for (int n = 0; n < n3; ++n) // Re( conj(Q3[n][m]) * A2[j][l][n] )
          re += Q3.q[n][m].re * A2[j][l][n].re + Q3.q[n][m].im * A2[j][l][n].im;
        out.v[j][l][m] = re;
        nrm += re * re;
      }
  double sc = csqrt(2.0 * l3 + 1.0) / csqrt(nrm);
  for (int j = 0; j < n1; ++j)
    for (int l = 0; l < n2; ++l)
      for (int m = 0; m < n3; ++m)
        out.v[j][l][m] *= sc;
  return out;
}

// One separate constexpr evaluation per path (keeps each under clang's step limit).
// NOTE: variables are WP<i> (not W<i>) to avoid colliding with the struct name W3.
#define TP_PATH_LIST(X) \
  X(0,0,0,0)  X(1,0,1,1)  X(2,0,2,2)  X(3,0,3,3) \
  X(4,1,0,1)  X(5,1,1,0)  X(6,1,1,2)  X(7,1,2,1)  X(8,1,2,3)  X(9,1,3,2) \
  X(10,2,0,2) X(11,2,1,1) X(12,2,1,3) X(13,2,2,0) X(14,2,2,2) X(15,2,3,1) X(16,2,3,3) \
  X(17,3,0,3) X(18,3,1,2) X(19,3,2,1) X(20,3,2,3) X(21,3,3,0) X(22,3,3,2)

#define TP_DECL_W(i,a,b,c) constexpr W3 WP##i = wigner3j(a,b,c);
TP_PATH_LIST(TP_DECL_W)
#undef TP_DECL_W

constexpr W3 getW(int i) {
  switch (i) {
#define TP_CASE_W(i,a,b,c) case i: return WP##i;
    TP_PATH_LIST(TP_CASE_W)
#undef TP_CASE_W
  }
  return WP0;
}

// Densified per-output-column coefficient tables.
// cc[n][k][j]: coefficient multiplying y[o2c[n]+j] contributing to G[k][n]; zero-padded.
struct alignas(32) Tables {
  float cc[KPAD][16][8];
  int   o2c[KPAD];
  int   posc[KPAD];
};

constexpr Tables buildTables() {
  Tables T{};
  // pos: stable argsort by l3
  int pos[NPATH] = {};
  {
    bool used[NPATH] = {};
    for (int ni = 0; ni < NPATH; ++ni) {
      int best = -1;
      for (int i = 0; i < NPATH; ++i)
        if (!used[i] && (best < 0 || PATH_L[i][2] < PATH_L[best][2])) best = i;
      used[best] = true;
      pos[best] = ni;
    }
  }
  // concat: list-insert simulation
  int concat[NPATH] = {};
  int len = 0;
  for (int idx = 0; idx < NPATH; ++idx) {
    int at = pos[idx] < len ? pos[idx] : len;
    for (int t = len; t > at; --t) concat[t] = concat[t - 1];
    concat[at] = idx;
    ++len;
  }
  // output column start per path
  int colStart[NPATH] = {};
  {
    int c = 0;
    for (int q = 0; q < NPATH; ++q) {
      int idx = concat[q];
      colStart[idx] = c;
      c += 2 * PATH_L[idx][2] + 1;
    }
  }
  for (int n = 0; n < KPAD; ++n) { T.o2c[n] = 0; T.posc[n] = 0; }
  for (int idx = 0; idx < NPATH; ++idx) {
    const int l1 = PATH_L[idx][0], l2 = PATH_L[idx][1], l3 = PATH_L[idx][2];
    const int o1 = OFFS[l1], o2 = OFFS[l2];
    const int w1 = 2 * l1 + 1, w2 = 2 * l2 + 1, w3 = 2 * l3 + 1;
    W3 wg = getW(idx);
    for (int kk = 0; kk < w3; ++kk) {
      const int n = colStart[idx] + kk;
      T.o2c[n] = o2;
      T.posc[n] = pos[idx];
      for (int k = 0; k < 16; ++k) {
        int a = k - o1;
        if (a >= 0 && a < w1)
          for (int j = 0; j < w2; ++j)
            T.cc[n][k][j] = (float)wg.v[a][j][kk];
      }
    }
  }
  return T;
}

__constant__ Tables g_T = buildTables();

// One wave32 per edge. out[e] (32x99) = x[e] (32x16) @ G (16x99), via
// v_wmma_f32_16x16x4_f32 chained over K, then scaled by weights.
__global__ __launch_bounds__(256) void tp_wmma_kernel(
    const float* __restrict__ x, const float* __restrict__ y,
    const float* __restrict__ w, float* __restrict__ out, int nedge) {
  const int lane = threadIdx.x & 31;
  const int wv   = threadIdx.x >> 5;
  const int m    = lane & 15;   // row (A) / column (B,D) within tile
  const int hs   = lane >> 4;   // half-select (K pair for A/B, M+8 for D)

  int e = blockIdx.x * 8 + wv;
  if (e >= nedge) e = nedge - 1;  // uniform clamp: duplicate waves redo identical work

  // ---- A: x[e] in WMMA f32 16x16x4 layout; v0/v1 = K pair (0,1) or (2,3) ----
  const float* xe = x + (size_t)e * (NCH * XD);
  v2f A[2][4];
#pragma unroll
  for (int mt = 0; mt < 2; ++mt)
#pragma unroll
    for (int ks = 0; ks < 4; ++ks)
      A[mt][ks] = __builtin_nontemporal_load(
          (const v2f*)(xe + (size_t)(mt * 16 + m) * XD + ks * 4 + 2 * hs));

  // ---- y: one value per lane, broadcast later via shuffles ----
  const float yreg = y[(size_t)e * XD + m];

  const float* we = w + (size_t)e * (NPATH * NCH);
  float* oe = out + (size_t)e * (NCH * KOUT);

  for (int nt = 0; nt < 7; ++nt) {
    const int n  = nt * 16 + m;      // output column (0..111; >=99 are pad)
    const int o2 = g_T.o2c[n];
    const int pw = g_T.posc[n];

    // broadcast the (up to 7, padded to 8) needed y values; full EXEC here
    float ys[8];
#pragma unroll
    for (int j = 0; j < 8; ++j) ys[j] = __shfl(yreg, (o2 + j) & 15, 32);

    // ---- B: G columns, built branchlessly from the densified tables ----
    v2f B[4];
#pragma unroll
    for (int ks = 0; ks < 4; ++ks) {
      float g[2];
#pragma unroll
      for (int t = 0; t < 2; ++t) {
        const int k = ks * 4 + 2 * hs + t;
        const v4f* cp = (const v4f*)(&g_T.cc[n][k][0]);
        const v4f c0 = cp[0], c1 = cp[1];
        g[t] = c0.x * ys[0] + c0.y * ys[1] + c0.z * ys[2] + c0.w * ys[3] +
               c1.x * ys[4] + c1.y * ys[5] + c1.z * ys[6] + c1.w * ys[7];
      }
      v2f b; b.x = g[0]; b.y = g[1];
      B[ks] = b;
    }

#pragma unroll
    for (int mt = 0; mt < 2; ++mt) {
      v8f acc = {0.f, 0.f, 0.f, 0.f, 0.f, 0.f, 0.f, 0.f};
#pragma unroll
      for (int ks = 0; ks < 4; ++ks)
        acc = __builtin_amdgcn_wmma_f32_16x16x4_f32(
            false, A[mt][ks], false, B[ks], (short)0, acc, false, false);

      // weight scale + store; D layout: VGPR r -> channel mt*16 + hs*8 + r, column n
      const int crow = mt * 16 + hs * 8;
      const float* wp = we + pw * NCH + crow;
      const v4f w0 = __builtin_nontemporal_load((const v4f*)(wp));
      const v4f w1 = __builtin_nontemporal_load((const v4f*)(wp + 4));
      const float warr[8] = {w0.x, w0.y, w0.z, w0.w, w1.x, w1.y, w1.z, w1.w};
      if (n < KOUT) {
        float* op = oe + (size_t)crow * KOUT + n;
#pragma unroll
        for (int r = 0; r < 8; ++r)
          __builtin_nontemporal_store(acc[r] * warr[r], op + (size_t)r * KOUT);
      }
    }
  }
}

} // namespace tp

extern "C" void kernel_launch(void* const* d_in, const int* in_sizes, int n_in,
                              void* d_out, int out_size, void* d_ws, size_t ws_size,
                              hipStream_t stream) {
  (void)n_in; (void)out_size; (void)d_ws; (void)ws_size;
  const float* x = (const float*)d_in[0];
  const float* y = (const float*)d_in[1];
  const float* w = (const float*)d_in[2];
  float* out = (float*)d_out;
  const int nedge = in_sizes[0] / (tp::NCH * tp::XD);
  if (nedge <= 0) return;
  const int blocks = (nedge + 7) / 8;  // 8 waves (edges) per 256-thread block
  tp::tp_wmma_kernel<<<blocks, 256, 0, stream>>>(x, y, w, out, nedge);
}